// CoreModel_38929583571044
// MI455X (gfx1250) — compile-verified
//
#include <hip/hip_runtime.h>

// ---------------------------------------------------------------------------
// Point-splat renderer for MI455X (gfx1250, wave32).
//
// d2(pixel p, point n) = (px-x)^2 + (py-y)^2 is an exact rank-4 product:
//   [px, py, px^2+py^2, 1] . [-2x, -2y, 1, x^2+y^2]
// -> V_WMMA_F32_16X16X4_F32 computes a 16-pixel x 16-point d2 tile per op.
//
// Each block renders exactly one image row (128 pixels, constant py), so a
// cooperative row-band cull first compacts the ~3-7% of points with
// |y_ndc - py| < R into an LDS index list (sentinel-padded to a tile
// boundary so the inner loop has no tail guards); the WMMA loop runs only
// over survivor tiles. Hits (d2 < R^2) go through a wave-wide __any vote
// into per-pixel LDS candidate lists; one thread per pixel depth-sorts
// (K=16), composites the 30-dim features front-to-back, applies SH-2,
// clips, and writes the row-flipped image.
// ---------------------------------------------------------------------------

#define SDIM   128
#define NPTS   4096
#define PTN    (NPTS + 16)   // point table incl. sentinel slots
#define KSEL   16
#define R2C    (0.03f * 0.03f)
#define FOCALC 2.0f
#define FEAT   30
#define CAP    48            // per-pixel candidate capacity (avg occupancy ~3)
#define MAXC   2048          // per-row survivor capacity (multiple of 32)

typedef float v2f __attribute__((ext_vector_type(2)));
typedef float v4f __attribute__((ext_vector_type(4)));
typedef float v8f __attribute__((ext_vector_type(8)));

// ---------------------------------------------------------------------------
// Kernel 1: project points, build packed WMMA B-fragment table.
// ws layout (floats): [0 .. 4*PTN)          PT[n] = (-2x, -2y, x^2+y^2, z)
//                     [4*PTN .. 4*PTN+N)    x_ndc
//                     [4*PTN+N .. 4*PTN+2N) y_ndc  (1e30 if z<=0: self-culls)
// PT[NPTS..] are sentinels with huge x^2+y^2 (tail padding for WMMA tiles).
// ---------------------------------------------------------------------------
__global__ void splat_prepare(const float* __restrict__ verts,
                              const float* __restrict__ camR,
                              const float* __restrict__ camT,
                              float* __restrict__ ws) {
    int n = blockIdx.x * blockDim.x + threadIdx.x;
    if (n >= NPTS) return;
    float vx = verts[n * 3 + 0];
    float vy = verts[n * 3 + 1];
    float vz = verts[n * 3 + 2];
    // cam = v @ R + T   (row-vector convention)
    float cx = vx * camR[0] + vy * camR[3] + vz * camR[6] + camT[0];
    float cy = vx * camR[1] + vy * camR[4] + vz * camR[7] + camT[1];
    float cz = vx * camR[2] + vy * camR[5] + vz * camR[8] + camT[2];
    float inv = 1.0f / cz;
    float xn = FOCALC * cx * inv;
    float yn = FOCALC * cy * inv;
    bool valid = (cz > 0.0f);
    v4f pt;
    pt.x = -2.0f * xn;
    pt.y = -2.0f * yn;
    pt.z = valid ? (xn * xn + yn * yn) : 3.0e38f;
    pt.w = cz;
    ((v4f*)ws)[n]          = pt;
    ws[4 * PTN + n]        = xn;
    ws[4 * PTN + NPTS + n] = valid ? yn : 1.0e30f;  // invalid -> fails band cull
    if (n < 16) {  // sentinel slots for tile tail padding
        v4f s; s.x = 0.0f; s.y = 0.0f; s.z = 3.0e38f; s.w = 1.0e30f;
        ((v4f*)ws)[NPTS + n] = s;
    }
}

// ---------------------------------------------------------------------------
// WMMA d2 tile: A = per-wave 16-pixel fragment, B from packed point float4.
// Lanes 0-15 carry K0/K1 rows, lanes 16-31 carry K2/K3 rows.
// ---------------------------------------------------------------------------
__device__ __forceinline__ v8f d2_tile(v2f a, v4f p, int hi) {
    v2f b;
    b.x = hi ? 1.0f : p.x;   // K2 row = 1        | K0 row = -2x
    b.y = hi ? p.z  : p.y;   // K3 row = x^2+y^2  | K1 row = -2y
    v8f c = {};
    return __builtin_amdgcn_wmma_f32_16x16x4_f32(
        /*neg_a=*/false, a, /*neg_b=*/false, b,
        /*c_mod=*/(short)0, c, /*reuse_a=*/false, /*reuse_b=*/false);
}

__device__ __forceinline__ void gather_hits(v8f c, float zval, int n, int mBase,
                                            int* scnt, float* sz, int* sid) {
    bool hit = false;
#pragma unroll
    for (int r = 0; r < 8; ++r) hit |= (c[r] < R2C);
    if (!__any(hit)) return;   // skip insertion when the wave-tile is empty
#pragma unroll
    for (int r = 0; r < 8; ++r) {
        if (c[r] < R2C) {
            int m = mBase + r;
            int slot = atomicAdd(&scnt[m], 1);
            if (slot < CAP) {
                sz [m * CAP + slot] = zval;
                sid[m * CAP + slot] = n;
            }
        }
    }
}

// ---------------------------------------------------------------------------
// Kernel 2: row-band cull + WMMA distance tiles + candidate gather +
// sort/composite/SH. Block = 256 threads (8 wave32) renders one image row;
// wave w owns columns [w*16, w*16+16). Grid = 128 blocks (one per row).
// ---------------------------------------------------------------------------
__global__ __launch_bounds__(256)
void splat_render(const float* __restrict__ ws,
                  const float* __restrict__ shp,       // N x 30 features
                  const float* __restrict__ viewdir,   // S x S x 3
                  float* __restrict__ out) {           // S x S x 3
    __shared__ int   scnt[128];
    __shared__ float sz   [128 * CAP];
    __shared__ int   sid  [128 * CAP];
    __shared__ int   plist[MAXC];
    __shared__ int   pcnt;

    const int tid  = threadIdx.x;
    const int lane = tid & 31;
    const int wave = tid >> 5;
    const int l16  = lane & 15;
    const int hi   = lane >> 4;      // 0: lanes 0-15 (K0/K1), 1: lanes 16-31 (K2/K3)

    if (tid < 128) scnt[tid] = 0;
    if (tid == 0)  pcnt = 0;
    __syncthreads();

    const v4f*   PT = (const v4f*)ws;
    const float* Xn = ws + 4 * PTN;
    const float* Yn = ws + 4 * PTN + NPTS;

    // This block = image row iy (output coords; flip folded: py = -g[iy]).
    const int   iy = blockIdx.x;
    const float py = -(1.0f - (2.0f * (float)iy + 1.0f) * (1.0f / SDIM));

    // ---- Row-band cull: keep points with (py - y_ndc)^2 < R^2.
    for (int n = tid; n < NPTS; n += 256) {
        float dy = py - Yn[n];
        if (dy * dy < R2C) {
            int slot = atomicAdd(&pcnt, 1);
            if (slot < MAXC) plist[slot] = n;
        }
    }
    __syncthreads();

    int cnt = pcnt;
    if (cnt > MAXC) cnt = MAXC;
    const int ntiles = ((cnt + 31) >> 5) << 1;   // even # of 16-point tiles
    const int padded = ntiles << 4;              // = ntiles*16 <= MAXC

    // Sentinel-pad the survivor list to the tile boundary (<= 31 entries),
    // so the WMMA loop needs no tail guards (unconditional ds_load).
    for (int i = cnt + tid; i < padded; i += 256) plist[i] = NPTS;
    __syncthreads();

    // ---- A fragment (loop invariant): row M = [px, py, px^2+py^2, 1].
    const int ixA = wave * 16 + l16;
    const float pxA = 1.0f - (2.0f * (float)ixA + 1.0f) * (1.0f / SDIM);
    v2f a;
    a.x = hi ? (pxA * pxA + py * py) : pxA;
    a.y = hi ? 1.0f                  : py;

    const int mBase = wave * 16 + 8 * hi;  // C/D rows this lane holds (M=r+8*hi)

    // ---- Survivor-tile loop (x2): gather B fragments by index, WMMA, insert.
    for (int t = 0; t < ntiles; t += 2) {
        const int n0 = plist[t * 16 + l16];
        const int n1 = plist[t * 16 + 16 + l16];
        v4f p0 = PT[n0];
        v4f p1 = PT[n1];

        v8f c0 = d2_tile(a, p0, hi);
        v8f c1 = d2_tile(a, p1, hi);

        gather_hits(c0, p0.w, n0, mBase, scnt, sz, sid);
        gather_hits(c1, p1.w, n1, mBase, scnt, sz, sid);
    }
    __syncthreads();

    // ---- Phase 2: one thread per pixel (column) of this row.
    if (tid < 128) {
        const int m = tid;
        int c2 = scnt[m];
        if (c2 > CAP) c2 = CAP;
        const int kk = c2 < KSEL ? c2 : KSEL;

        float* zl = &sz [m * CAP];
        int*   id = &sid[m * CAP];
        // Partial selection sort: front kk entries = kk smallest depths.
        for (int i = 0; i < kk; ++i) {
            int best = i;
            float bz = zl[i];
            for (int j = i + 1; j < c2; ++j)
                if (zl[j] < bz) { bz = zl[j]; best = j; }
            float tz = zl[i]; zl[i] = zl[best]; zl[best] = tz;
            int   ti = id[i]; id[i] = id[best]; id[best] = ti;
        }

        const int ix = m;
        const float px = 1.0f - (2.0f * (float)ix + 1.0f) * (1.0f / SDIM);

        float acc[FEAT];
#pragma unroll
        for (int c = 0; c < FEAT; ++c) acc[c] = 0.0f;

        float T = 1.0f;
        for (int i = 0; i < kk; ++i) {
            const int n = id[i];
            const float dx = px - Xn[n];
            const float dy = py - Yn[n];
            const float d2 = dx * dx + dy * dy;
            const float w = 1.0f - d2 * (1.0f / R2C);
            const float contrib = w * T;
            T *= (1.0f - w);
            const float* f = shp + (long)n * FEAT;
#pragma unroll
            for (int c = 0; c < FEAT; ++c) acc[c] += contrib * f[c];
        }

        // SH-2 evaluation with normalized view direction.
        const int vbase = (iy * SDIM + ix) * 3;
        const float vx = viewdir[vbase + 0];
        const float vy = viewdir[vbase + 1];
        const float vz = viewdir[vbase + 2];
        const float inv = rsqrtf(vx * vx + vy * vy + vz * vz);
        const float x = vx * inv, y = vy * inv, z = vz * inv;

        const float C0  = 0.28209479177387814f;
        const float C1  = 0.4886025119029199f;
        const float C20 =  1.0925484305920792f;
        const float C21 = -1.0925484305920792f;
        const float C22 =  0.31539156525252005f;
        const float C23 = -1.0925484305920792f;
        const float C24 =  0.5462742152960396f;

        const float b0 = C0;
        const float b1 = -C1 * y;
        const float b2 =  C1 * z;
        const float b3 = -C1 * x;
        const float b4 = C20 * x * y;
        const float b5 = C21 * y * z;
        const float b6 = C22 * (2.0f * z * z - x * x - y * y);
        const float b7 = C23 * x * z;
        const float b8 = C24 * (x * x - y * y);

#pragma unroll
        for (int c = 0; c < 3; ++c) {
            float r = acc[c]
                    + b0 * acc[3 + 0 * 3 + c] + b1 * acc[3 + 1 * 3 + c]
                    + b2 * acc[3 + 2 * 3 + c] + b3 * acc[3 + 3 * 3 + c]
                    + b4 * acc[3 + 4 * 3 + c] + b5 * acc[3 + 5 * 3 + c]
                    + b6 * acc[3 + 6 * 3 + c] + b7 * acc[3 + 7 * 3 + c]
                    + b8 * acc[3 + 8 * 3 + c];
            r = fminf(fmaxf(r, 0.0f), 1.0f);
            out[vbase + c] = r;
        }
    }
}

// ---------------------------------------------------------------------------
extern "C" void kernel_launch(void* const* d_in, const int* in_sizes, int n_in,
                              void* d_out, int out_size, void* d_ws, size_t ws_size,
                              hipStream_t stream) {
    (void)in_sizes; (void)n_in; (void)out_size; (void)ws_size;
    const float* verts   = (const float*)d_in[0];  // N x 3
    const float* shp     = (const float*)d_in[1];  // N x 30
    const float* viewdir = (const float*)d_in[2];  // S x S x 3
    const float* camR    = (const float*)d_in[3];  // 3 x 3
    const float* camT    = (const float*)d_in[4];  // 3
    float* out = (float*)d_out;
    float* ws  = (float*)d_ws;                     // needs 4*PTN + 2*N floats < 97 KB

    splat_prepare<<<(NPTS + 255) / 256, 256, 0, stream>>>(verts, camR, camT, ws);
    splat_render<<<SDIM, 256, 0, stream>>>(ws, shp, viewdir, out);
}